// LocalConcatSheafLearnerVariant_55628416418072
// MI455X (gfx1250) — compile-verified
//
#include <hip/hip_runtime.h>

// Problem constants (from the reference)
#define N_ROWS   60000      // nodes*D rows of x
#define HID      128
#define E_EDGES  960000
#define GROUPS   320000     // E/3
#define MTILES   3750       // N_ROWS/16
#define YSTRIDE  32         // padded projected row: [0..8]=y_row, [16..24]=y_col, rest exact zeros

typedef __attribute__((ext_vector_type(2))) float v2f;
typedef __attribute__((ext_vector_type(8))) float v8f;

// ---------------------------------------------------------------------------
// Kernel 1: per-node projection  y[n, 0..8] = x[n] @ W[:, :128]^T
//                                y[n,16..24] = x[n] @ W[:,128:]^T
// via V_WMMA_F32_16X16X4_F32, one wave per 16-row tile, K=128 in 32 steps.
// ---------------------------------------------------------------------------
__global__ __launch_bounds__(256) void sheaf_proj_kernel(
    const float* __restrict__ x,   // [N_ROWS, HID]
    const float* __restrict__ W,   // [9, 256]
    float* __restrict__ y)         // [N_ROWS, YSTRIDE] workspace
{
    // LDS-transposed weights: L[k][j]   = W[j][k]       (j<9, row part)
    //                         L[k][10+j]= W[j][128+k]   (j<9, col part)
    // padded to 26 so masked lane reads (n up to 15) stay in bounds.
    __shared__ float L[128][26];
    const int tid = threadIdx.x;

    for (int i = tid; i < 128 * 26; i += 256) ((float*)L)[i] = 0.0f;
    __syncthreads();
    for (int i = tid; i < 9 * 256; i += 256) {
        const int j = i >> 8;          // output channel 0..8
        const int k = i & 255;         // input channel 0..255
        const float w = W[j * 256 + k];
        if (k < 128) L[k][j]            = w;
        else         L[k - 128][10 + j] = w;
    }
    __syncthreads();

    const int lane  = tid & 31;
    const int wave  = tid >> 5;
    const int mtile = blockIdx.x * 8 + wave;   // wave-uniform
    if (mtile >= MTILES) return;               // whole-wave exit: EXEC stays all-ones for WMMA

    const int   n   = lane & 15;               // B-matrix column / A-matrix row within tile
    const int   hi  = lane >> 4;               // lane half selects K phase
    const float msk = (n < 9) ? 1.0f : 0.0f;   // zero pad cols 9..15 without EXEC divergence

    const float* xrow = x + (long)(mtile * 16 + n) * HID;

    v8f c0, c1;
    #pragma unroll
    for (int r = 0; r < 8; ++r) { c0[r] = 0.0f; c1[r] = 0.0f; }

    #pragma unroll 4
    for (int kk = 0; kk < 32; ++kk) {
        const int kb = kk * 4 + hi * 2;        // this lane's K pair (ISA 16x4 f32 A layout)
        v2f a;  a.x  = xrow[kb];               a.y  = xrow[kb + 1];
        v2f b0; b0.x = L[kb][n]      * msk;    b0.y = L[kb + 1][n]      * msk;
        v2f b1; b1.x = L[kb][10 + n] * msk;    b1.y = L[kb + 1][10 + n] * msk;
        // D = A(16x4,f32) x B(4x16,f32) + C(16x16,f32)
        c0 = __builtin_amdgcn_wmma_f32_16x16x4_f32(false, a, false, b0, (short)0, c0, false, false);
        c1 = __builtin_amdgcn_wmma_f32_16x16x4_f32(false, a, false, b1, (short)0, c1, false, false);
    }

    // C/D layout: VGPR r, lanes 0-15 -> M=r, lanes 16-31 -> M=r+8; N = lane&15
    float* ybase = y + (long)mtile * 16 * YSTRIDE;
    #pragma unroll
    for (int r = 0; r < 8; ++r) {
        const int orow = r + hi * 8;
        ybase[orow * YSTRIDE + n]      = c0[r];
        ybase[orow * YSTRIDE + 16 + n] = c1[r];
    }
}

// ---------------------------------------------------------------------------
// Kernel 2: per-group gather of projected rows (L2-resident), sum, tanh.
// out[g, o] = tanh( sum_d  y[row[3g+d], o] + y[col[3g+d], 16+o] ),  o=0..8
// ---------------------------------------------------------------------------
__global__ __launch_bounds__(256) void sheaf_gather_kernel(
    const float* __restrict__ y,   // [N_ROWS, YSTRIDE]
    const int*   __restrict__ ei,  // [2, E] row-major: [0..E)=row, [E..2E)=col
    float* __restrict__ out)       // [GROUPS, 9]
{
    const int g = blockIdx.x * 256 + threadIdx.x;
    if (g >= GROUPS) return;

    float4 a0 = make_float4(0.f, 0.f, 0.f, 0.f);
    float4 a1 = a0, a2 = a0;

    #pragma unroll
    for (int d = 0; d < 3; ++d) {
        const int e = 3 * g + d;
        const int r = ei[e];
        const int c = ei[E_EDGES + e];
        const float4* yr = reinterpret_cast<const float4*>(y + (long)r * YSTRIDE);      // cols 0..11 (9..11 == 0)
        const float4* yc = reinterpret_cast<const float4*>(y + (long)c * YSTRIDE + 16); // cols 16..27 (25..27 == 0)
        const float4 r0 = yr[0], r1 = yr[1], r2 = yr[2];
        const float4 q0 = yc[0], q1 = yc[1], q2 = yc[2];
        a0.x += r0.x + q0.x; a0.y += r0.y + q0.y; a0.z += r0.z + q0.z; a0.w += r0.w + q0.w;
        a1.x += r1.x + q1.x; a1.y += r1.y + q1.y; a1.z += r1.z + q1.z; a1.w += r1.w + q1.w;
        a2.x += r2.x + q2.x; a2.y += r2.y + q2.y; a2.z += r2.z + q2.z; a2.w += r2.w + q2.w;
    }

    float* o = out + (long)g * 9;
    o[0] = tanhf(a0.x); o[1] = tanhf(a0.y); o[2] = tanhf(a0.z);
    o[3] = tanhf(a0.w); o[4] = tanhf(a1.x); o[5] = tanhf(a1.y);
    o[6] = tanhf(a1.z); o[7] = tanhf(a1.w); o[8] = tanhf(a2.x);
}

extern "C" void kernel_launch(void* const* d_in, const int* in_sizes, int n_in,
                              void* d_out, int out_size, void* d_ws, size_t ws_size,
                              hipStream_t stream) {
    const float* x  = (const float*)d_in[0];   // [60000, 128] f32
    const float* W  = (const float*)d_in[1];   // [9, 256] f32
    const int*   ei = (const int*)d_in[2];     // [2, 960000] (integer input -> const int*)
    float*       out = (float*)d_out;          // [320000, 3, 3] f32
    float*       y   = (float*)d_ws;           // needs N_ROWS*YSTRIDE*4 = 7.68 MB scratch

    sheaf_proj_kernel<<<(MTILES + 7) / 8, 256, 0, stream>>>(x, W, y);
    sheaf_gather_kernel<<<(GROUPS + 255) / 256, 256, 0, stream>>>(y, ei, out);
}